// BatchSuperQMulti_12463995093547
// MI455X (gfx1250) — compile-verified
//
#include <hip/hip_runtime.h>
#include <hip/hip_bf16.h>
#include <math.h>

namespace {
constexpr int   Bn = 16, Nn = 64, Mn = 4096, Kn = 7168;
constexpr int   Pn = Mn + Kn;          // 11264 points per batch
constexpr int   PS = 24;               // per-primitive param stride (floats)
constexpr float MIN_S = 0.01f, MIN_E = 0.1f, MAX_E = 1.9f;
constexpr float TRUNC = 0.05f, EPSF = 1e-6f;
constexpr float LOG2E = 1.4426950408889634f;

typedef __attribute__((ext_vector_type(2))) float v2f;
typedef __attribute__((ext_vector_type(8))) float v8f;

__device__ __forceinline__ float scl(float v) {          // signed_clamp
  float s = (v > 0.f) ? 1.f : -1.f;
  return s * fmaxf(fabsf(v), EPSF);
}
// b in [1e-6, finite] -> raw v_log_f32 / v_exp_f32 are safe (no fixup code)
__device__ __forceinline__ float fpow(float b, float p) {
  return __builtin_amdgcn_exp2f(__builtin_amdgcn_logf(b) * p);
}
__device__ __forceinline__ float fexp(float x) {         // exp(x), x <= 0
  return __builtin_amdgcn_exp2f(x * LOG2E);
}
} // namespace

__global__ __launch_bounds__(256) void superq_sdf_kernel(
    const float* __restrict__ raw_scale, const float* __restrict__ raw_exp,
    const float* __restrict__ raw_rot,   const float* __restrict__ trans,
    const float* __restrict__ raw_taper, const float* __restrict__ pts,
    const float* __restrict__ opts,      const unsigned char* __restrict__ mask8,
    float* __restrict__ out)
{
  __shared__ float sp[Nn * PS];   // 6 KB: 64 primitives x 24 floats
  const int b   = blockIdx.y;
  const int tid = threadIdx.x;

  // ---- per-primitive parameter prep (threads 0..63), written to LDS ----
  if (tid < Nn) {
    const int i = b * Nn + tid;
    float qw = raw_rot[i*4+0], qx = raw_rot[i*4+1];
    float qy = raw_rot[i*4+2], qz = raw_rot[i*4+3];
    float rn = rsqrtf(qw*qw + qx*qx + qy*qy + qz*qz);
    qw *= rn; qx *= rn; qy *= rn; qz *= rn;
    float* o = sp + tid * PS;
    o[0]=1.f-2.f*(qy*qy+qz*qz); o[1]=2.f*(qx*qy-qw*qz); o[2]=2.f*(qx*qz+qw*qy);
    o[3]=2.f*(qx*qy+qw*qz); o[4]=1.f-2.f*(qx*qx+qz*qz); o[5]=2.f*(qy*qz-qw*qx);
    o[6]=2.f*(qx*qz-qw*qy); o[7]=2.f*(qy*qz+qw*qx); o[8]=1.f-2.f*(qx*qx+qy*qy);
    o[9]=trans[i*3+0]; o[10]=trans[i*3+1]; o[11]=trans[i*3+2];
    float sx = expf(raw_scale[i*3+0]) + MIN_S;
    float sy = expf(raw_scale[i*3+1]) + MIN_S;
    float sz = expf(raw_scale[i*3+2]) + MIN_S;
    o[12]=1.f/sx; o[13]=1.f/sy; o[14]=1.f/sz;
    float e1 = 1.f/(1.f+expf(-raw_exp[i*2+0])) * (MAX_E-MIN_E) + MIN_E;
    float e2 = 1.f/(1.f+expf(-raw_exp[i*2+1])) * (MAX_E-MIN_E) + MIN_E;
    o[15]=1.f/e1; o[16]=1.f/e2; o[17]=e2/e1; o[18]=-0.5f*e1;
    o[19]=tanhf(raw_taper[i*2+0])/sz; o[20]=tanhf(raw_taper[i*2+1])/sz;
    o[21]=mask8[i] ? 1.f : 0.f;
  }
  __syncthreads();

  // ---- per-point work: each wave owns 32 consecutive points ----
  const int  lane = tid & 31;
  const int  l    = lane & 15;
  const bool hi   = lane >= 16;
  const int  wv   = tid >> 5;
  const int  gp   = blockIdx.x * blockDim.x + wv * 32 + lane;  // point idx in batch

  float px, py, pz;
  if (gp < Mn) {                       // 4096 % 32 == 0 -> wave-uniform branch
    const float* s = pts + ((size_t)b * Mn + gp) * 3;
    px = s[0]; py = s[1]; pz = s[2];
  } else {
    const float* s = opts + ((size_t)b * Kn + (gp - Mn)) * 3;
    px = s[0]; py = s[1]; pz = s[2];
  }
  // Cross-lane staging for the two 16-point WMMA tiles (loop-invariant).
  float pz0 = __shfl(pz, l, 32);        // Z of point (lane&15)       [tile0, hi half]
  float px1 = __shfl(px, l | 16, 32);   // X of point (lane&15)+16    [tile1, lo half]
  float py1 = __shfl(py, l | 16, 32);   // Y of point (lane&15)+16    [tile1, lo half]
  float pa0 = hi ? pz0 : px;
  float pa1 = hi ? pz  : px1;
  const bool l3 = (l < 3);

  float mrun = -__builtin_inff();       // streaming softmax state
  float den = 0.f, num = 0.f;

  for (int n = 0; n < Nn; ++n) {
    const float* pm = sp + n * PS;
    // wave-uniform mask skip (scalar branch -> EXEC stays all-1s at WMMA)
    if (__builtin_amdgcn_readfirstlane(__float_as_int(pm[21])) == 0) continue;

    const float tx = pm[9], ty = pm[10], tz = pm[11];
    // A = R^T padded to 16x4 in the ISA A-layout (VGPR0:K0/K2, VGPR1:K1/K3).
    // Gather addresses are in-bounds for every lane (max idx 21 < 24), so load
    // unconditionally and select -> no divergent EXEC region around the WMMAs.
    float ga0 = pm[(hi ? 6 : 0) + l];
    float ga1 = pm[3 + l];
    float a0 = l3 ? ga0 : 0.f;
    float a1 = (l3 && !hi) ? ga1 : 0.f;
    v2f A = {a0, a1};
    const float tsel = hi ? tz : tx;
    v2f B0 = {pa0 - tsel, hi ? 0.f : (py  - ty)};   // tile0: points base..base+15
    v2f B1 = {pa1 - tsel, hi ? 0.f : (py1 - ty)};   // tile1: points base+16..+31
    v8f zc = {};
    v8f d0 = __builtin_amdgcn_wmma_f32_16x16x4_f32(false, A, false, B0,
                                                   (short)0, zc, false, false);
    v8f d1 = __builtin_amdgcn_wmma_f32_16x16x4_f32(false, A, false, B1,
                                                   (short)0, zc, false, false);
    // D rows 0..2 (lanes 0-15) = rotated x,y,z; merge tile1 into hi lanes.
    float sx1 = __shfl(d1[0], l, 32);   // executed by all lanes (convergent)
    float sy1 = __shfl(d1[1], l, 32);
    float sz1 = __shfl(d1[2], l, 32);
    float X = hi ? sx1 : d0[0];
    float Y = hi ? sy1 : d0[1];
    float Z = hi ? sz1 : d0[2];

    float x = scl(X), y = scl(Y), z = scl(Z);
    float fx = scl(pm[19]*z + 1.f);     // kx/sz * z + 1
    float fy = scl(pm[20]*z + 1.f);
    x *= __builtin_amdgcn_rcpf(fx);     // |fx| >= 1e-6 -> approx rcp is safe
    y *= __builtin_amdgcn_rcpf(fy);
    float r0 = __builtin_amdgcn_sqrtf(x*x + y*y + z*z);
    float t1 = fmaxf(fabsf(x*pm[12]), EPSF); t1 = fmaxf(t1*t1, EPSF);
    float t2 = fmaxf(fabsf(y*pm[13]), EPSF); t2 = fmaxf(t2*t2, EPSF);
    float t3 = fmaxf(fabsf(z*pm[14]), EPSF); t3 = fmaxf(t3*t3, EPSF);
    float term1 = fpow(t1, pm[16]);     // ^(1/e2)
    float term2 = fpow(t2, pm[16]);
    float term3 = fpow(t3, pm[15]);     // ^(1/e1)
    float g = fpow(fmaxf(term1 + term2, EPSF), pm[17]);  // ^(e2/e1)
    float f = fpow(fmaxf(g + term3, EPSF), pm[18]);      // ^(-e1/2)
    float sdf = r0 * (1.f - f);

    float clip  = fminf(fmaxf(sdf, -TRUNC), TRUNC);
    float leaky = clip + 0.1f * (sdf - clip);
    float logit = -100.f * sdf;
    // online softmax accumulate (args always <= 0 -> raw v_exp safe)
    float nm = fmaxf(mrun, logit);
    float cs = fexp(mrun  - nm);
    float ce = fexp(logit - nm);
    den = den * cs + ce;
    num = num * cs + ce * leaky;
    mrun = nm;
  }
  out[(size_t)b * Pn + gp] = num / den;
}

extern "C" void kernel_launch(void* const* d_in, const int* in_sizes, int n_in,
                              void* d_out, int out_size, void* d_ws, size_t ws_size,
                              hipStream_t stream) {
  (void)in_sizes; (void)n_in; (void)d_ws; (void)ws_size; (void)out_size;
  const float* raw_scale = (const float*)d_in[0];
  const float* raw_exp   = (const float*)d_in[1];
  const float* raw_rot   = (const float*)d_in[2];
  const float* trans     = (const float*)d_in[3];
  const float* raw_taper = (const float*)d_in[4];
  const float* pts       = (const float*)d_in[5];
  const float* opts      = (const float*)d_in[6];
  const unsigned char* m = (const unsigned char*)d_in[7];
  float* out = (float*)d_out;
  dim3 grid(Pn / 256, Bn, 1);   // 44 x 16 blocks, 256 threads (8 wave32)
  superq_sdf_kernel<<<grid, 256, 0, stream>>>(raw_scale, raw_exp, raw_rot, trans,
                                              raw_taper, pts, opts, m, out);
}